// LightGCNModel_29695403884795
// MI455X (gfx1250) — compile-verified
//
#include <hip/hip_runtime.h>
#include <stdint.h>

#define N_NODES 100000
#define EMBED   64
#define N_EDGES 1200000
#define ALPHA   0.25f

__device__ __forceinline__ void atomic_add_f32(float* p, float v) {
  (void)__hip_atomic_fetch_add(p, v, __ATOMIC_RELAXED, __HIP_MEMORY_SCOPE_AGENT);
}

__global__ void zero_f32_kernel(float* __restrict__ p, int n) {
  int i = blockIdx.x * blockDim.x + threadIdx.x;
  if (i < n) p[i] = 0.0f;
}

// deg[col] += edge_attr * mask   (masked-out edges contribute 0 -> skip atomic)
__global__ void deg_kernel(const float* __restrict__ attr,
                           const unsigned char* __restrict__ mask,
                           const long long* __restrict__ col,
                           float* __restrict__ deg) {
  int e = blockIdx.x * blockDim.x + threadIdx.x;
  if (e < N_EDGES) {
    float w = mask[e] ? attr[e] : 0.0f;
    if (w != 0.0f) atomic_add_f32(&deg[(int)col[e]], w);
  }
}

__global__ void dinv_kernel(float* __restrict__ deg) {
  int i = blockIdx.x * blockDim.x + threadIdx.x;
  if (i < N_NODES) {
    float d = deg[i];
    deg[i] = (d > 0.0f) ? rsqrtf(d) : 0.0f;
  }
}

// norm[e] = dinv[row] * w * dinv[col]; computed once, reused by all 3 layers
__global__ void norm_kernel(const float* __restrict__ attr,
                            const unsigned char* __restrict__ mask,
                            const long long* __restrict__ row,
                            const long long* __restrict__ col,
                            const float* __restrict__ dinv,
                            float* __restrict__ norm) {
  int e = blockIdx.x * blockDim.x + threadIdx.x;
  if (e < N_EDGES) {
    float w = mask[e] ? attr[e] : 0.0f;
    norm[e] = dinv[(int)row[e]] * w * dinv[(int)col[e]];
  }
}

// out = ALPHA*x ; hA = x ; hB = 0   (float4-vectorized, B128 traffic)
__global__ void init_kernel(const float4* __restrict__ x,
                            float4* __restrict__ out,
                            float4* __restrict__ hA,
                            float4* __restrict__ hB, int n4) {
  int i = blockIdx.x * blockDim.x + threadIdx.x;
  if (i < n4) {
    float4 v = x[i];
    float4 o; o.x = ALPHA * v.x; o.y = ALPHA * v.y; o.z = ALPHA * v.z; o.w = ALPHA * v.w;
    out[i] = o;
    hA[i] = v;
    float4 z; z.x = z.y = z.z = z.w = 0.0f;
    hB[i] = z;
  }
}

// out += ALPHA * hsum ; hzero = 0 (becomes next layer's accumulator)
__global__ void finish_kernel(float4* __restrict__ out,
                              const float4* __restrict__ hsum,
                              float4* __restrict__ hzero, int n4) {
  int i = blockIdx.x * blockDim.x + threadIdx.x;
  if (i < n4) {
    float4 h = hsum[i];
    float4 o = out[i];
    o.x += ALPHA * h.x; o.y += ALPHA * h.y; o.z += ALPHA * h.z; o.w += ALPHA * h.w;
    out[i] = o;
    float4 z; z.x = z.y = z.z = z.w = 0.0f;
    hzero[i] = z;
  }
}

#define PROP_BLOCK 256
#define WAVES_PER_BLOCK (PROP_BLOCK / 32)
#define DEPTH 4  // async pipeline depth (LDS buffers per wave)

// One wave32 = 2 edges per step: 16 lanes x 16B = one 256B feature row.
// Feature rows are gathered with gfx1250 async load-to-LDS into a per-wave
// ring of 4 buffers; ASYNCcnt pipelining keeps 4 random gathers in flight
// while the global_atomic_add_f32 scatter of the oldest pair executes.
// Per-wave edge ranges are contiguous so norm/row/col metadata streams hit
// the same cachelines across iterations. Out-of-range pipeline slots are
// padded with a safe dummy gather (norm==0 -> consumer skips) so the
// issue/consume FIFO stays balanced and control flow stays wave-uniform
// (async ops ignore EXEC; counter discipline must be exact).
__global__ __launch_bounds__(PROP_BLOCK)
void propagate_kernel(const float* __restrict__ h_in,
                      float* __restrict__ h_out,
                      const float* __restrict__ norm,
                      const long long* __restrict__ row,
                      const long long* __restrict__ col) {
  __shared__ float4 stage[WAVES_PER_BLOCK * DEPTH * 32];  // 16 KB

  const int lane = threadIdx.x & 31;
  const int wave = threadIdx.x >> 5;
  const int half = lane >> 4;   // which edge of the pair this lane serves
  const int sub  = lane & 15;   // 16-byte chunk within the 256B row

  const int nPairs = N_EDGES / 2;
  const int waveId = blockIdx.x * WAVES_PER_BLOCK + wave;
  const int nWaves = gridDim.x * WAVES_PER_BLOCK;

  // contiguous pair range for this wave
  const int start = (int)((long long)waveId * nPairs / nWaves);
  const int end   = (int)((long long)(waveId + 1) * nPairs / nWaves);
  const int cnt   = end - start;

  // LDS byte address of this lane's slot in buffer 0; buffer b = +b*512
  const uint32_t lds_lane = (uint32_t)(uintptr_t)(&stage[0]) +
                            (uint32_t)(((wave * DEPTH) * 32 + lane) * 16);

  float nrm0, nrm1, nrm2, nrm3;
  int   c0, c1, c2, c3;

  auto issue = [&](int k, int buf, float& nrm_o, int& col_o) {
    float nrm = 0.0f;
    int   c   = 0;
    uint32_t goff = (uint32_t)sub * 16u;  // pad slot: safe gather of h_in[0..]
    if (k < cnt) {
      int e = (start + k) * 2 + half;
      nrm  = norm[e];
      c    = (int)col[e];
      goff = (uint32_t)(int)row[e] * (uint32_t)(EMBED * 4) + (uint32_t)sub * 16u;
    }
    asm volatile("global_load_async_to_lds_b128 %0, %1, %2"
                 :: "v"(lds_lane + (uint32_t)buf * 512u), "v"(goff), "s"(h_in)
                 : "memory");
    nrm_o = nrm;
    col_o = c;
  };

  auto consume = [&](int buf, float nrm, int c) {
    float4 v = stage[(wave * DEPTH + buf) * 32 + lane];
    if (nrm != 0.0f) {
      float* dst = h_out + (size_t)c * EMBED + sub * 4;
      atomic_add_f32(dst + 0, nrm * v.x);
      atomic_add_f32(dst + 1, nrm * v.y);
      atomic_add_f32(dst + 2, nrm * v.z);
      atomic_add_f32(dst + 3, nrm * v.w);
    }
  };

  // prologue: fill the 4-deep ring
  issue(0, 0, nrm0, c0);
  issue(1, 1, nrm1, c1);
  issue(2, 2, nrm2, c2);
  issue(3, 3, nrm3, c3);

  // steady state: consume oldest, refill its buffer (FIFO, 4 in flight)
  const int Kp = ((cnt + DEPTH - 1) / DEPTH) * DEPTH;  // padded trip count
  for (int k = DEPTH; k < Kp; k += DEPTH) {
    asm volatile("s_wait_asynccnt 0x3" ::: "memory");
    consume(0, nrm0, c0); issue(k + 0, 0, nrm0, c0);
    asm volatile("s_wait_asynccnt 0x3" ::: "memory");
    consume(1, nrm1, c1); issue(k + 1, 1, nrm1, c1);
    asm volatile("s_wait_asynccnt 0x3" ::: "memory");
    consume(2, nrm2, c2); issue(k + 2, 2, nrm2, c2);
    asm volatile("s_wait_asynccnt 0x3" ::: "memory");
    consume(3, nrm3, c3); issue(k + 3, 3, nrm3, c3);
  }

  // drain
  asm volatile("s_wait_asynccnt 0x3" ::: "memory"); consume(0, nrm0, c0);
  asm volatile("s_wait_asynccnt 0x2" ::: "memory"); consume(1, nrm1, c1);
  asm volatile("s_wait_asynccnt 0x1" ::: "memory"); consume(2, nrm2, c2);
  asm volatile("s_wait_asynccnt 0x0" ::: "memory"); consume(3, nrm3, c3);
}

extern "C" void kernel_launch(void* const* d_in, const int* in_sizes, int n_in,
                              void* d_out, int out_size, void* d_ws, size_t ws_size,
                              hipStream_t stream) {
  (void)in_sizes; (void)n_in; (void)out_size; (void)ws_size;

  const float*         x    = (const float*)d_in[0];
  const float*         attr = (const float*)d_in[1];
  const long long*     eidx = (const long long*)d_in[2];     // int64 [2, E]
  const unsigned char* mask = (const unsigned char*)d_in[3]; // bool, 1 byte

  const long long* row = eidx;             // source
  const long long* col = eidx + N_EDGES;   // target
  float* out = (float*)d_out;

  // workspace layout (floats): norm[E] | deg/dinv[N] | hA[N*64] | hB[N*64]
  float* ws    = (float*)d_ws;
  float* normw = ws;
  float* deg   = ws + N_EDGES;
  size_t off   = ((size_t)N_EDGES + (size_t)N_NODES + 63) & ~(size_t)63;
  float* hA    = ws + off;
  float* hB    = hA + (size_t)N_NODES * EMBED;

  const int n4 = N_NODES * EMBED / 4;
  const int B  = 256;

  zero_f32_kernel<<<(N_NODES + B - 1) / B, B, 0, stream>>>(deg, N_NODES);
  deg_kernel<<<(N_EDGES + B - 1) / B, B, 0, stream>>>(attr, mask, col, deg);
  dinv_kernel<<<(N_NODES + B - 1) / B, B, 0, stream>>>(deg);
  norm_kernel<<<(N_EDGES + B - 1) / B, B, 0, stream>>>(attr, mask, row, col, deg, normw);
  init_kernel<<<(n4 + B - 1) / B, B, 0, stream>>>((const float4*)x, (float4*)out,
                                                  (float4*)hA, (float4*)hB, n4);

  const int PG = 2048;  // 16384 waves; ~37 contiguous edge-pairs per wave
  // layer 1: hA -> hB
  propagate_kernel<<<PG, PROP_BLOCK, 0, stream>>>(hA, hB, normw, row, col);
  finish_kernel<<<(n4 + B - 1) / B, B, 0, stream>>>((float4*)out, (const float4*)hB,
                                                    (float4*)hA, n4);
  // layer 2: hB -> hA
  propagate_kernel<<<PG, PROP_BLOCK, 0, stream>>>(hB, hA, normw, row, col);
  finish_kernel<<<(n4 + B - 1) / B, B, 0, stream>>>((float4*)out, (const float4*)hA,
                                                    (float4*)hB, n4);
  // layer 3: hA -> hB
  propagate_kernel<<<PG, PROP_BLOCK, 0, stream>>>(hA, hB, normw, row, col);
  finish_kernel<<<(n4 + B - 1) / B, B, 0, stream>>>((float4*)out, (const float4*)hB,
                                                    (float4*)hA, n4);
}